// AttentionLayer_2680059592730
// MI455X (gfx1250) — compile-verified
//
#include <hip/hip_runtime.h>
#include <hip/hip_bf16.h>
#include <math.h>

typedef __attribute__((ext_vector_type(2))) float v2f;
typedef __attribute__((ext_vector_type(8))) float v8f;

#define B_   64
#define T_   2048
#define DENC 512
#define DRNN 1024
#define DATT 128
#define NF_  32
#define K_   31

// Branch-free tanh: 1 - 2/(exp(2x)+1), native exp/rcp, saturates to +/-1.
__device__ __forceinline__ float tanh_fast(float x) {
    float e = __expf(2.0f * x);                 // v_exp_f32 path, no branches
    return 1.0f - 2.0f * __builtin_amdgcn_rcpf(e + 1.0f);
}

// ---------------------------------------------------------------------------
// Kernel 1: processed query  pq[b,a] = sum_d h[b,d] * qw[a,d]
// ---------------------------------------------------------------------------
__global__ void att_pq(const float* __restrict__ h,
                       const float* __restrict__ qw,
                       float* __restrict__ pq) {
    __shared__ float h_s[DRNN];
    const int b = blockIdx.x, tid = threadIdx.x;
    for (int i = tid; i < DRNN; i += 128) h_s[i] = h[b * DRNN + i];
    __syncthreads();
    const float* w = &qw[tid * DRNN];
    float acc = 0.f;
    #pragma unroll 8
    for (int d = 0; d < DRNN; ++d) acc += h_s[d] * w[d];
    pq[b * DATT + tid] = acc;   // tid in [0,128)
}

// ---------------------------------------------------------------------------
// Kernel 2: fused location-conv + loc-dense (FP32 WMMA) + tanh energies + dot
//   grid (32 t-tiles, 64 b), block 128 (4 waves). Each wave owns 16 t-rows.
// ---------------------------------------------------------------------------
__global__ void att_energies(const float* __restrict__ pm,     // [B,T,128]
                             const float* __restrict__ awcat,  // [B,2,T]
                             const unsigned char* __restrict__ mask, // [B,T]
                             const float* __restrict__ pq,     // [B,128]
                             const float* __restrict__ vw,     // [128]
                             const float* __restrict__ wconv,  // [32,2,31]
                             const float* __restrict__ ldw,    // [128,32]
                             float* __restrict__ align_out) {  // [B,T]
    __shared__ float aw_s[2 * (64 + 30)];     // padded input slice
    __shared__ float wc_s[NF_ * 2 * K_];      // conv weights
    __shared__ float loc_s[64 * NF_];         // conv output tile [t_local][f]
    __shared__ float pq_s[DATT];
    __shared__ float vw_s[DATT];
    __shared__ float align_s[64];

    const int tid = threadIdx.x;
    const int b  = blockIdx.y;
    const int t0 = blockIdx.x * 64;

    // stage inputs
    for (int i = tid; i < 2 * 94; i += 128) {
        const int c = i / 94, j = i - c * 94;
        const int t = t0 - 15 + j;
        aw_s[i] = (t >= 0 && t < T_) ? awcat[(b * 2 + c) * T_ + t] : 0.f;
    }
    for (int i = tid; i < NF_ * 2 * K_; i += 128) wc_s[i] = wconv[i];
    pq_s[tid] = pq[b * DATT + tid];
    vw_s[tid] = vw[tid];
    __syncthreads();

    // 1D conv (2 -> 32 channels, 'same'), 16 (t,f) entries per thread
    #pragma unroll
    for (int e = 0; e < 16; ++e) {
        const int idx = e * 128 + tid;
        const int tl = idx >> 5, f = idx & 31;
        const float* w0 = &wc_s[f * 2 * K_];
        const float* a0 = &aw_s[tl];
        const float* a1 = &aw_s[94 + tl];
        float acc = 0.f;
        #pragma unroll
        for (int k = 0; k < K_; ++k)
            acc = fmaf(a0[k], w0[k], fmaf(a1[k], w0[K_ + k], acc));
        loc_s[idx] = acc;
    }
    __syncthreads();

    // loc-dense via V_WMMA_F32_16X16X4_F32, fused with energies.
    const int wv   = tid >> 5;        // wave id 0..3 -> t rows [wv*16, wv*16+16)
    const int lane = tid & 31;
    const int m    = lane & 15;       // A: row M / B,D: column N
    const int hi   = lane >> 4;       // K sub-pair select for A/B; M-half for D
    const int trowA = wv * 16 + m;

    float asum[8];
    #pragma unroll
    for (int r = 0; r < 8; ++r) asum[r] = 0.f;

    for (int nt = 0; nt < 8; ++nt) {          // N tiles: a = nt*16 + m
        v8f acc = {};
        const int a_col = nt * 16 + m;
        #pragma unroll
        for (int kk = 0; kk < 8; ++kk) {      // K = 32 in steps of 4
            const int klo = kk * 4 + hi * 2;
            v2f av, bv;
            av.x = loc_s[trowA * NF_ + klo];
            av.y = loc_s[trowA * NF_ + klo + 1];
            bv.x = ldw[a_col * NF_ + klo];    // B[k][n] = ldw[n][k]
            bv.y = ldw[a_col * NF_ + klo + 1];
            acc = __builtin_amdgcn_wmma_f32_16x16x4_f32(
                false, av, false, bv, (short)0, acc, false, false);
        }
        const float vwa = vw_s[a_col];
        const float pqa = pq_s[a_col];
        #pragma unroll
        for (int r = 0; r < 8; ++r) {
            const int trr = wv * 16 + hi * 8 + r;     // D-row -> t_local
            const float x = pqa + acc[r] +
                pm[((size_t)(b * T_ + t0 + trr)) * DATT + a_col];
            asum[r] = fmaf(vwa, tanh_fast(x), asum[r]);
        }
    }

    // reduce the 16 lanes of each half (same t row, different a columns)
    #pragma unroll
    for (int r = 0; r < 8; ++r) {
        float s = asum[r];
        s += __shfl_xor(s, 1, 32);
        s += __shfl_xor(s, 2, 32);
        s += __shfl_xor(s, 4, 32);
        s += __shfl_xor(s, 8, 32);
        if (m == 0) align_s[wv * 16 + hi * 8 + r] = s;
    }
    __syncthreads();

    if (tid < 64) {
        const int t = t0 + tid;
        float v = align_s[tid];
        if (mask[b * T_ + t]) v = -__builtin_huge_valf();
        align_out[(size_t)b * T_ + t] = v;
    }
}

// ---------------------------------------------------------------------------
// Kernel 3: in-place softmax over T per batch row. grid B_, block 256.
// ---------------------------------------------------------------------------
__global__ void att_softmax(float* __restrict__ wts) {
    __shared__ float red[256];
    const int b = blockIdx.x, tid = threadIdx.x;
    float* row = wts + (size_t)b * T_;
    float vals[8];
    float mx = -__builtin_huge_valf();
    #pragma unroll
    for (int i = 0; i < 8; ++i) { vals[i] = row[tid + i * 256]; mx = fmaxf(mx, vals[i]); }
    red[tid] = mx; __syncthreads();
    for (int s = 128; s > 0; s >>= 1) {
        if (tid < s) red[tid] = fmaxf(red[tid], red[tid + s]);
        __syncthreads();
    }
    mx = red[0]; __syncthreads();
    float sum = 0.f;
    #pragma unroll
    for (int i = 0; i < 8; ++i) { vals[i] = __expf(vals[i] - mx); sum += vals[i]; }
    red[tid] = sum; __syncthreads();
    for (int s = 128; s > 0; s >>= 1) {
        if (tid < s) red[tid] += red[tid + s];
        __syncthreads();
    }
    const float inv = 1.f / red[0];
    #pragma unroll
    for (int i = 0; i < 8; ++i) row[tid + i * 256] = vals[i] * inv;
}

// ---------------------------------------------------------------------------
// Kernel 4a: context partials over t-chunks (deterministic two-stage reduce)
//   grid (8 t-chunks, 2 d-chunks, 64 b), block 256. Streams 256 MB `memory`.
// ---------------------------------------------------------------------------
__global__ void ctx_partial(const float* __restrict__ mem,  // [B,T,512]
                            const float* __restrict__ wts,  // [B,T]
                            float* __restrict__ part) {     // [B,2,8,256]
    __shared__ float w_s[256];
    const int tc = blockIdx.x, dc = blockIdx.y, b = blockIdx.z, tid = threadIdx.x;
    w_s[tid] = wts[(size_t)b * T_ + tc * 256 + tid];
    __syncthreads();
    const float* mrow = mem + ((size_t)b * T_ + tc * 256) * DENC + dc * 256 + tid;
    float acc = 0.f;
    for (int t = 0; t < 256; t += 8) {
        // prefetch 64 rows ahead (block-uniform guard -> scalar branch only)
        if (t < 192)
            __builtin_prefetch(mrow + (size_t)(t + 64) * DENC, 0, 0);
        #pragma unroll
        for (int u = 0; u < 8; ++u)
            acc = fmaf(w_s[t + u], mrow[(size_t)(t + u) * DENC], acc);
    }
    part[(((b * 2 + dc) * 8) + tc) * 256 + tid] = acc;
}

// Kernel 4b: fixed-order reduce of the 8 partials. grid B_*2, block 256.
__global__ void ctx_reduce(const float* __restrict__ part,
                           float* __restrict__ ctx) {
    const int dc = blockIdx.x & 1, b = blockIdx.x >> 1, tid = threadIdx.x;
    float acc = 0.f;
    #pragma unroll
    for (int tc = 0; tc < 8; ++tc)
        acc += part[(((b * 2 + dc) * 8) + tc) * 256 + tid];
    ctx[b * DENC + dc * 256 + tid] = acc;
}

// ---------------------------------------------------------------------------
extern "C" void kernel_launch(void* const* d_in, const int* in_sizes, int n_in,
                              void* d_out, int out_size, void* d_ws, size_t ws_size,
                              hipStream_t stream) {
    const float* h     = (const float*)d_in[0];           // [B, DRNN]
    const float* mem   = (const float*)d_in[1];           // [B, T, DENC]
    const float* pm    = (const float*)d_in[2];           // [B, T, DATT]
    const float* awcat = (const float*)d_in[3];           // [B, 2, T]
    const unsigned char* mask = (const unsigned char*)d_in[4]; // [B, T] bool
    const float* qw    = (const float*)d_in[5];           // [DATT, DRNN]
    const float* vw    = (const float*)d_in[6];           // [1, DATT]
    const float* wconv = (const float*)d_in[7];           // [NF, 2, K]
    const float* ldw   = (const float*)d_in[8];           // [DATT, NF]

    float* ctx_out = (float*)d_out;                       // [B, DENC]
    float* wts_out = (float*)d_out + (size_t)B_ * DENC;   // [B, T]

    float* pq_ws   = (float*)d_ws;                        // B*DATT floats (32 KB)
    float* part_ws = (float*)((char*)d_ws + (size_t)B_ * DATT * sizeof(float)); // 1 MB

    att_pq<<<dim3(B_), dim3(128), 0, stream>>>(h, qw, pq_ws);
    att_energies<<<dim3(T_ / 64, B_), dim3(128), 0, stream>>>(
        pm, awcat, mask, pq_ws, vw, wconv, ldw, wts_out);
    att_softmax<<<dim3(B_), dim3(256), 0, stream>>>(wts_out);
    ctx_partial<<<dim3(8, 2, B_), dim3(256), 0, stream>>>(mem, wts_out, part_ws);
    ctx_reduce<<<dim3(B_ * 2), dim3(256), 0, stream>>>(part_ws, ctx_out);
}